// SpatialCorrelationSampler_25701084299497
// MI455X (gfx1250) — compile-verified
//
#include <hip/hip_runtime.h>

typedef __attribute__((ext_vector_type(2))) float v2f;
typedef __attribute__((ext_vector_type(8))) float v8f;

typedef __attribute__((address_space(1))) int gas_int;   // global
typedef __attribute__((address_space(3))) int las_int;   // LDS

#define B_    8
#define C_    256
#define H_    64
#define W_    96
#define P_    21
#define PAD_  20
#define KC    16                    // channels per LDS chunk
#define NCH   (C_ / KC)             // 16 chunks
#define NJ    48                    // stride-2 in2 columns staged per row
#define A_FLOATS (KC * 16)          // 256 floats
#define B_FLOATS (P_ * KC * NJ)     // 16128 floats
#define BUF_FLOATS (A_FLOATS + B_FLOATS)   // 16384 floats = 64 KB
#define LDS_FLOATS (2 * BUF_FLOATS)        // 128 KB double buffer

#if __has_builtin(__builtin_amdgcn_global_load_async_to_lds_b32)
#define USE_ASYNC 1
#else
#define USE_ASYNC 0
#endif

__global__ __launch_bounds__(256)
void corr_wmma_f32(const float* __restrict__ in1,
                   const float* __restrict__ in2,
                   float* __restrict__ out)
{
    extern __shared__ float smem[];
    const int tid  = threadIdx.x;
    const int lane = tid & 31;
    const int w    = tid >> 5;      // wave id (8 waves)
    const int kh   = lane >> 4;     // K-half (selects channel pair)
    const int mn   = lane & 15;     // M (A) / N (B) index

    const int bid    = blockIdx.x;               // 0 .. 3071
    const int g3     = bid % 6;
    const int yy     = (bid / 6) % H_;
    const int b      = bid / (6 * H_);
    const int parity = g3 & 1;
    const int g      = g3 >> 1;
    const int xb     = parity + 32 * g;          // output x = xb + 2*m

    // stager decomposition: fixed (j-low, channel) per thread, loop py
    const int sj = tid & 15;        // j = sj + 16*jh
    const int sc = tid >> 4;        // channel within chunk, 0..15

    // wave-uniform valid py range (rows outside stay pre-zeroed = padding)
    int pyLo = (21 - yy) >> 1; if (pyLo < 0)  pyLo = 0;
    int pyHi = (83 - yy) >> 1; if (pyHi > 20) pyHi = 20;

    // this wave's 8 (py, band-tile) assignments; tile 63 is a clamped dummy
    int pyA[8], tA[8];
#pragma unroll
    for (int i = 0; i < 8; ++i) {
        int tile = w * 8 + i;
        int py   = tile / 3;
        pyA[i]   = (py > 20) ? 20 : py;
        tA[i]    = tile % 3;
    }

    v8f acc[8];
#pragma unroll
    for (int i = 0; i < 8; ++i) acc[i] = (v8f)0.0f;

    // ---- pre-zero LDS once: OOB (py,j) slots are channel-independent and are
    //      never overwritten, so they provide the zero-padding for all chunks.
    for (int e = tid; e < LDS_FLOATS; e += 256) smem[e] = 0.0f;
    __syncthreads();

    // ---------------- staging helpers ----------------
    auto stageA = [&](int chunk, float* buf) {
        // one element per thread: c = sc, m = sj (always in-bounds)
        const float* ga = in1 + (((b * C_) + chunk * KC + sc) * H_ + yy) * W_
                              + xb + 2 * sj;
        float* la = buf + ((sc >> 1) * 16 + sj) * 2 + (sc & 1);
#if USE_ASYNC
        __builtin_amdgcn_global_load_async_to_lds_b32(
            (gas_int*)ga, (las_int*)la, 0, 0);
#else
        *la = *ga;
#endif
    };

    auto stageB = [&](int chunk, float* buf) {
        const float* gbase = in2 + (((b * C_) + chunk * KC + sc) * H_
                                    + (yy - PAD_)) * W_ + (xb - PAD_);
        float* lbase = buf + A_FLOATS + (sc >> 1) * (NJ * 2) + (sc & 1);
#pragma unroll
        for (int jh = 0; jh < 3; ++jh) {
            const int j  = sj + 16 * jh;
            const int co = xb + 2 * j - PAD_;
            if ((unsigned)co < (unsigned)W_) {       // lane predicate, once per jh
                const float* ga = gbase + 2 * j + 2 * pyLo * W_;
                float*       la = lbase + j * 2 + pyLo * (KC / 2 * NJ * 2);
                for (int py = pyLo; py <= pyHi; ++py) {   // unpredicated body
#if USE_ASYNC
                    __builtin_amdgcn_global_load_async_to_lds_b32(
                        (gas_int*)ga, (las_int*)la, 0, 0);
#else
                    *la = *ga;
#endif
                    ga += 2 * W_;
                    la += (KC / 2 * NJ * 2);         // 768 floats per py
                }
            }
        }
    };

    // ---------------- compute: fragment-pipelined WMMA ----------------
    auto compute = [&](const float* buf) {
        const float* As = buf;
        const float* Bs = buf + A_FLOATS;
        auto ldA = [&](int kk) -> v2f {
            int cp = 2 * kk + kh;
            return *(const v2f*)(As + (cp * 16 + mn) * 2);
        };
        auto ldB = [&](int kk, int i) -> v2f {
            int cp = 2 * kk + kh;
            return *(const v2f*)(Bs + ((pyA[i] * (KC / 2) + cp) * NJ
                                       + 16 * tA[i] + mn) * 2);
        };
        v2f aF[2];
        v2f bF[2][8];
        aF[0] = ldA(0);
#pragma unroll
        for (int i = 0; i < 8; ++i) bF[0][i] = ldB(0, i);
#pragma unroll
        for (int kk = 0; kk < KC / 4; ++kk) {
            const int cur = kk & 1, nxt = cur ^ 1;
            if (kk + 1 < KC / 4) {            // prefetch next K-step fragments
                aF[nxt] = ldA(kk + 1);
#pragma unroll
                for (int i = 0; i < 8; ++i) bF[nxt][i] = ldB(kk + 1, i);
            }
#pragma unroll
            for (int i = 0; i < 8; ++i)
                acc[i] = __builtin_amdgcn_wmma_f32_16x16x4_f32(
                    false, aF[cur], false, bF[cur][i],
                    (short)0, acc[i], false, false);
        }
    };

    // ---------------- main pipeline ----------------
    float* buf0 = smem;
    float* buf1 = smem + BUF_FLOATS;

    stageA(0, buf0);
    stageB(0, buf0);
#if USE_ASYNC
    asm volatile("s_wait_asynccnt 0x0" ::: "memory");
#endif
    __syncthreads();

    for (int k = 0; k < NCH; ++k) {
        float* cur = (k & 1) ? buf1 : buf0;
        float* nxt = (k & 1) ? buf0 : buf1;
        if (k + 1 < NCH) {                    // overlap staging with compute
            stageA(k + 1, nxt);
            stageB(k + 1, nxt);
        }
        compute(cur);
#if USE_ASYNC
        asm volatile("s_wait_asynccnt 0x0" ::: "memory");
#endif
        __syncthreads();
    }

    // ---------------- transpose each 16x16 tile via LDS, store ----------------
    float* stage = smem + w * (16 * 17);      // 17-float pitch, conflict-free
    const float scale = 1.0f / (float)C_;

#pragma unroll
    for (int i = 0; i < 8; ++i) {
        const int tile = w * 8 + i;
#pragma unroll
        for (int r = 0; r < 8; ++r)           // D layout: M = r + 8*kh, N = mn
            stage[(r + 8 * kh) * 17 + mn] = acc[i][r];
        asm volatile("s_wait_dscnt 0x0" ::: "memory");

        if (tile < 63) {
            const int py = pyA[i];
            const int T  = tA[i];
#pragma unroll
            for (int r2 = 0; r2 < 8; ++r2) {  // lane owns M = mn, walks N
                int n2 = r2 + 8 * kh;
                int px = 16 * T + n2 - mn;    // band condition
                if ((unsigned)px <= 20u) {
                    float v = stage[mn * 17 + n2] * scale;
                    out[(((b * (P_ * P_)) + py * P_ + px) * H_ + yy) * W_
                        + xb + 2 * mn] = v;
                }
            }
        }
        asm volatile("s_wait_dscnt 0x0" ::: "memory");
    }
}

extern "C" void kernel_launch(void* const* d_in, const int* in_sizes, int n_in,
                              void* d_out, int out_size, void* d_ws, size_t ws_size,
                              hipStream_t stream)
{
    (void)in_sizes; (void)n_in; (void)d_ws; (void)ws_size; (void)out_size;
    const float* in1 = (const float*)d_in[0];
    const float* in2 = (const float*)d_in[1];
    float* out = (float*)d_out;

    dim3 grid(B_ * H_ * 6);      // (b, y, x-parity, x-group) = 3072 blocks
    dim3 block(256);             // 8 wave32s
    size_t shmem = (size_t)LDS_FLOATS * sizeof(float);   // 128 KB
    corr_wmma_f32<<<grid, block, shmem, stream>>>(in1, in2, out);
}